// Net_3092376453218
// MI455X (gfx1250) — compile-verified
//
#include <hip/hip_runtime.h>
#include <hip/hip_bf16.h>

// ---------------- problem constants (match reference) ----------------
#define BGRAPH 200
#define NPG1   512
#define EDGES  1638400          // B * 8192
#define FINC   11
#define HDIM   128
#define K1     256              // ceil(0.5*512)
#define K2     205              // ceil(0.8*256)
#define K3     164              // ceil(0.8*205)
#define N1     (BGRAPH*NPG1)    // 102400 nodes after layer 1 conv
#define N2     (BGRAPH*K1)      // 51200 nodes after pool 1
#define N3     (BGRAPH*K2)      // 41000 nodes after pool 2
#define N4     (BGRAPH*K3)      // 32800 nodes after pool 3

typedef __attribute__((ext_vector_type(2))) float v2f;
typedef __attribute__((ext_vector_type(8))) float v8f;

// ---------------- utility ----------------
__global__ void zero_kernel(float* __restrict__ p, int n) {
  int i = blockIdx.x * blockDim.x + threadIdx.x;
  if (i < n) p[i] = 0.f;
}

__global__ void init_edges_kernel(const int* __restrict__ es, const int* __restrict__ ed,
                                  int* __restrict__ sw, int* __restrict__ dw,
                                  float* __restrict__ mw, int E) {
  int e = blockIdx.x * blockDim.x + threadIdx.x;
  if (e >= E) return;
  sw[e] = es[e]; dw[e] = ed[e]; mw[e] = 1.f;
}

// ---------------- layer 1: aggregate in FIN=11 space (linear => pre-transform sum) ----
__global__ void agg1_kernel(const float* __restrict__ x, const int* __restrict__ src,
                            const int* __restrict__ dst, float* __restrict__ agg, int E) {
  int e = blockIdx.x * blockDim.x + threadIdx.x;
  if (e >= E) return;
  const float* xr = x   + (size_t)src[e] * FINC;
  float*       ar = agg + (size_t)dst[e] * FINC;
#pragma unroll
  for (int k = 0; k < FINC; ++k) atomicAdd(&ar[k], xr[k]);
}

__global__ void conv1_kernel(const float* __restrict__ x, const float* __restrict__ agg,
                             const float* __restrict__ Wrel, const float* __restrict__ Wroot,
                             const float* __restrict__ bias, float* __restrict__ h, int M) {
  int idx = blockIdx.x * blockDim.x + threadIdx.x;
  if (idx >= M * HDIM) return;
  int nid = idx >> 7, cch = idx & 127;
  const float* ar = agg  + (size_t)nid * FINC;
  const float* xr = x    + (size_t)nid * FINC;
  const float* w1 = Wrel  + (size_t)cch * FINC;
  const float* w2 = Wroot + (size_t)cch * FINC;
  float s = bias[cch];
#pragma unroll
  for (int k = 0; k < FINC; ++k) s += ar[k] * w1[k] + xr[k] * w2[k];
  h[idx] = fmaxf(s, 0.f);
}

// ---------------- 128-wide masked aggregation (layers 2/3) ----------------
__global__ void aggH_kernel(const float* __restrict__ h, const int* __restrict__ src,
                            const int* __restrict__ dst, const float* __restrict__ mask,
                            float* __restrict__ agg, int E) {
  int idx = blockIdx.x * blockDim.x + threadIdx.x;
  int e = idx >> 5;
  if (e >= E) return;
  if (mask[e] == 0.f) return;                       // pooled-out edge
  int c = (idx & 31) * 4;
  const float4 v = *(const float4*)(h + (size_t)src[e] * HDIM + c);
  float* a = agg + (size_t)dst[e] * HDIM + c;
  atomicAdd(a + 0, v.x); atomicAdd(a + 1, v.y);
  atomicAdd(a + 2, v.z); atomicAdd(a + 3, v.w);
}

// ---------------- dual-GEMM with f32 WMMA: D = act(A@W^T [+ A2@W2^T] + bias) ---------
// One wave32 per 16x16 tile; K stepped by 4 using V_WMMA_F32_16X16X4_F32.
// Per ISA layout: lanes 0-15 carry K={kk,kk+1}, lanes 16-31 carry K={kk+2,kk+3};
// A rows indexed by lane, B (=W^T) columns indexed by lane -> both are contiguous
// 8-byte loads from row-major A and row-major W.
template<bool DUAL, bool RELU>
__global__ void gemm_wmma_f32(const float* __restrict__ A, const float* __restrict__ A2,
                              const float* __restrict__ W, const float* __restrict__ W2,
                              const float* __restrict__ bias, float* __restrict__ D,
                              int M, int N, int K) {
  const int tilesN = N >> 4;
  const int tilesM = (M + 15) >> 4;
  int wave = (blockIdx.x * blockDim.x + threadIdx.x) >> 5;
  if (wave >= tilesM * tilesN) return;
  int tm = wave / tilesN, tn = wave - tm * tilesN;
  int lane = threadIdx.x & 31;
  int half = lane >> 4;       // 0: K+{0,1}, 1: K+{2,3}
  int l    = lane & 15;
  int m0 = tm << 4, n0 = tn << 4;
  int arow = m0 + l; if (arow >= M) arow = M - 1;   // clamp; extra rows never stored
  int wrow = n0 + l;
  const float* Ap  = A + (size_t)arow * K + half * 2;
  const float* Wp  = W + (size_t)wrow * K + half * 2;
  const float* A2p = DUAL ? (A2 + (size_t)arow * K + half * 2) : nullptr;
  const float* W2p = DUAL ? (W2 + (size_t)wrow * K + half * 2) : nullptr;
  v8f c = {0.f, 0.f, 0.f, 0.f, 0.f, 0.f, 0.f, 0.f};
  for (int kk = 0; kk < K; kk += 4) {
    if ((kk & 28) == 0) __builtin_prefetch(Ap + kk + 32, 0, 1);  // global_prefetch_b8
    v2f a = *(const v2f*)(Ap + kk);
    v2f b = *(const v2f*)(Wp + kk);
    c = __builtin_amdgcn_wmma_f32_16x16x4_f32(false, a, false, b, (short)0, c, false, false);
    if (DUAL) {
      v2f a2 = *(const v2f*)(A2p + kk);
      v2f b2 = *(const v2f*)(W2p + kk);
      c = __builtin_amdgcn_wmma_f32_16x16x4_f32(false, a2, false, b2, (short)0, c, false, false);
    }
  }
  int col = n0 + l;
  float bb = bias[col];
#pragma unroll
  for (int r = 0; r < 8; ++r) {
    int row = m0 + r + half * 8;                    // C/D layout: vgpr r -> M=r (+8 for hi lanes)
    if (row < M) {
      float v = c[r] + bb;
      if (RELU) v = fmaxf(v, 0.f);
      D[(size_t)row * N + col] = v;
    }
  }
}

// ---------------- top-k pooling pieces ----------------
__global__ void pwnorm_kernel(const float* __restrict__ pw1, const float* __restrict__ pw2,
                              const float* __restrict__ pw3, float* __restrict__ invn) {
  __shared__ float red[128];
  const float* pw = (blockIdx.x == 0) ? pw1 : (blockIdx.x == 1) ? pw2 : pw3;
  int t = threadIdx.x;
  float v = pw[t];
  red[t] = v * v;
  __syncthreads();
  for (int s = 64; s > 0; s >>= 1) { if (t < s) red[t] += red[t + s]; __syncthreads(); }
  if (t == 0) invn[blockIdx.x] = rsqrtf(red[0]);
}

__global__ void score_kernel(const float* __restrict__ h, const float* __restrict__ pw,
                             const float* __restrict__ invn, int which,
                             float* __restrict__ score, int n) {
  int i = blockIdx.x * blockDim.x + threadIdx.x;
  if (i >= n) return;
  const float4* hr = (const float4*)(h + (size_t)i * HDIM);
  const float4* pr = (const float4*)pw;
  float s = 0.f;
#pragma unroll
  for (int k = 0; k < HDIM / 4; ++k) {
    float4 a = hr[k], b = pr[k];
    s += a.x * b.x + a.y * b.y + a.z * b.z + a.w * b.w;
  }
  score[i] = tanhf(s * invn[which]);
}

// Per-graph descending bitonic sort of (score,idx) in LDS; emits perm + newid.
// blockDim.x = NPAD (power of two >= N, <= 512). Scores are in (-1,1); -2 pads.
__global__ void topk_kernel(const float* __restrict__ score, int* __restrict__ perm,
                            int* __restrict__ newid, int N, int k) {
  __shared__ float sc[512];
  __shared__ int   si[512];
  const int NPAD = blockDim.x;
  int g = blockIdx.x, t = threadIdx.x;
  sc[t] = (t < N) ? score[g * N + t] : -2.0f;
  si[t] = t;
  if (t < N) newid[g * N + t] = -1;
  __syncthreads();
  for (int size = 2; size <= NPAD; size <<= 1) {
    for (int stride = size >> 1; stride > 0; stride >>= 1) {
      int j = t ^ stride;
      if (j > t) {
        bool swapit = ((t & size) == 0) ? (sc[t] < sc[j])   // descending block
                                        : (sc[t] > sc[j]);
        if (swapit) {
          float tf = sc[t]; sc[t] = sc[j]; sc[j] = tf;
          int   ti = si[t]; si[t] = si[j]; si[j] = ti;
        }
      }
      __syncthreads();
    }
  }
  if (t < k) {
    int local = si[t];
    perm[g * k + t]      = g * N + local;
    newid[g * N + local] = g * k + t;
  }
}

__global__ void pool_gather_kernel(const float* __restrict__ h, const float* __restrict__ score,
                                   const int* __restrict__ perm, float* __restrict__ hp, int Nk) {
  int idx = blockIdx.x * blockDim.x + threadIdx.x;
  if (idx >= Nk * HDIM) return;
  int n = idx >> 7, f = idx & 127;
  int p = perm[n];
  hp[idx] = h[(size_t)p * HDIM + f] * score[p];
}

__global__ void remap_edges_kernel(int* __restrict__ src, int* __restrict__ dst,
                                   float* __restrict__ mask, const int* __restrict__ newid, int E) {
  int e = blockIdx.x * blockDim.x + threadIdx.x;
  if (e >= E) return;
  int ns = newid[src[e]], nd = newid[dst[e]];
  float m = mask[e];
  if (ns < 0 || nd < 0) m = 0.f;
  src[e] = ns < 0 ? 0 : ns;
  dst[e] = nd < 0 ? 0 : nd;
  mask[e] = m;
}

// ---------------- readout: cat(max, mean) over each graph; z += or z = ----------------
__global__ void readout_kernel(const float* __restrict__ hp, float* __restrict__ z, int k, int acc) {
  int g = blockIdx.x, f = threadIdx.x;  // 128 threads
  const float* base = hp + (size_t)g * k * HDIM + f;
  float mx = -3.4e38f, sm = 0.f;
  for (int n = 0; n < k; ++n) {
    float v = base[(size_t)n * HDIM];
    mx = fmaxf(mx, v); sm += v;
  }
  float mean = sm / (float)k;
  float* zg = z + (size_t)g * 256;
  if (acc) { zg[f] += mx; zg[128 + f] += mean; }
  else     { zg[f]  = mx; zg[128 + f]  = mean; }
}

// ---------------- final 64 -> 1 linear ----------------
__global__ void mlp_final_kernel(const float* __restrict__ z2, const float* __restrict__ Wl3,
                                 const float* __restrict__ bl3, float* __restrict__ out) {
  int g = blockIdx.x * blockDim.x + threadIdx.x;
  if (g >= BGRAPH) return;
  float s = bl3[0];
#pragma unroll
  for (int k = 0; k < 64; ++k) s += z2[(size_t)g * 64 + k] * Wl3[k];
  out[g] = s;
}

// ---------------------------------------------------------------------------
extern "C" void kernel_launch(void* const* d_in, const int* in_sizes, int n_in,
                              void* d_out, int out_size, void* d_ws, size_t ws_size,
                              hipStream_t stream) {
  (void)in_sizes; (void)n_in; (void)out_size; (void)ws_size;

  const float* x     = (const float*)d_in[0];
  const int*   esrc  = (const int*)  d_in[1];
  const int*   edst  = (const int*)  d_in[2];
  // d_in[3]=num_graphs, d_in[4]=nodes_per_graph (compile-time constants here)
  const float* Wrel1 = (const float*)d_in[5];
  const float* Wroot1= (const float*)d_in[6];
  const float* b1    = (const float*)d_in[7];
  const float* pw1   = (const float*)d_in[8];
  const float* Wrel2 = (const float*)d_in[9];
  const float* Wroot2= (const float*)d_in[10];
  const float* b2    = (const float*)d_in[11];
  const float* pw2   = (const float*)d_in[12];
  const float* Wrel3 = (const float*)d_in[13];
  const float* Wroot3= (const float*)d_in[14];
  const float* b3    = (const float*)d_in[15];
  const float* pw3   = (const float*)d_in[16];
  const float* Wl1   = (const float*)d_in[17];
  const float* bl1   = (const float*)d_in[18];
  const float* Wl2   = (const float*)d_in[19];
  const float* bl2   = (const float*)d_in[20];
  const float* Wl3   = (const float*)d_in[21];
  const float* bl3   = (const float*)d_in[22];
  float* out = (float*)d_out;

  // ---- workspace carve (bump allocator, 256B aligned) ----
  char* p = (char*)d_ws;
  auto carve = [&](size_t bytes) -> void* {
    void* r = (void*)p;
    p += (bytes + 255) & ~(size_t)255;
    return r;
  };
  float* agg   = (float*)carve((size_t)N2 * HDIM * 4);   // also holds agg1 (N1*11 < N2*128)
  float* h     = (float*)carve((size_t)N1 * HDIM * 4);   // conv outputs (reused each layer)
  float* hp    = (float*)carve((size_t)N2 * HDIM * 4);   // pooled features (reused)
  float* score = (float*)carve((size_t)N1 * 4);
  int*   perm  = (int*)  carve((size_t)N2 * 4);
  int*   newid = (int*)  carve((size_t)N1 * 4);
  int*   srcw  = (int*)  carve((size_t)EDGES * 4);
  int*   dstw  = (int*)  carve((size_t)EDGES * 4);
  float* maskw = (float*)carve((size_t)EDGES * 4);
  float* invn  = (float*)carve(16);
  float* z     = (float*)carve((size_t)BGRAPH * 256 * 4);
  float* z1v   = (float*)carve((size_t)BGRAPH * 512 * 4);
  float* z2v   = (float*)carve((size_t)BGRAPH * 64 * 4);

  auto cdiv = [](int a, int b) { return (a + b - 1) / b; };
  auto gemm_grid = [&](int M, int N) { return cdiv(((M + 15) / 16) * (N / 16) * 32, 256); };

  // ---- working edge arrays + pw norms ----
  init_edges_kernel<<<cdiv(EDGES, 256), 256, 0, stream>>>(esrc, edst, srcw, dstw, maskw, EDGES);
  pwnorm_kernel<<<3, 128, 0, stream>>>(pw1, pw2, pw3, invn);

  // ---- layer 1: conv (aggregate in FIN space, then fused dual linear + relu) ----
  zero_kernel<<<cdiv(N1 * FINC, 256), 256, 0, stream>>>(agg, N1 * FINC);
  agg1_kernel<<<cdiv(EDGES, 256), 256, 0, stream>>>(x, srcw, dstw, agg, EDGES);
  conv1_kernel<<<cdiv(N1 * HDIM, 256), 256, 0, stream>>>(x, agg, Wrel1, Wroot1, b1, h, N1);

  // ---- pool 1 (ratio 0.5: 512 -> 256) + readout x1 ----
  score_kernel<<<cdiv(N1, 256), 256, 0, stream>>>(h, pw1, invn, 0, score, N1);
  topk_kernel<<<BGRAPH, 512, 0, stream>>>(score, perm, newid, NPG1, K1);
  pool_gather_kernel<<<cdiv(N2 * HDIM, 256), 256, 0, stream>>>(h, score, perm, hp, N2);
  remap_edges_kernel<<<cdiv(EDGES, 256), 256, 0, stream>>>(srcw, dstw, maskw, newid, EDGES);
  readout_kernel<<<BGRAPH, 128, 0, stream>>>(hp, z, K1, 0);

  // ---- layer 2: conv (masked 128-wide aggregation, WMMA dual GEMM) ----
  zero_kernel<<<cdiv(N2 * HDIM, 256), 256, 0, stream>>>(agg, N2 * HDIM);
  aggH_kernel<<<cdiv(EDGES * 32, 256), 256, 0, stream>>>(hp, srcw, dstw, maskw, agg, EDGES);
  gemm_wmma_f32<true, true><<<gemm_grid(N2, HDIM), 256, 0, stream>>>(
      agg, hp, Wrel2, Wroot2, b2, h, N2, HDIM, HDIM);

  // ---- pool 2 (ratio 0.8: 256 -> 205) + readout x2 ----
  score_kernel<<<cdiv(N2, 256), 256, 0, stream>>>(h, pw2, invn, 1, score, N2);
  topk_kernel<<<BGRAPH, 256, 0, stream>>>(score, perm, newid, K1, K2);
  pool_gather_kernel<<<cdiv(N3 * HDIM, 256), 256, 0, stream>>>(h, score, perm, hp, N3);
  remap_edges_kernel<<<cdiv(EDGES, 256), 256, 0, stream>>>(srcw, dstw, maskw, newid, EDGES);
  readout_kernel<<<BGRAPH, 128, 0, stream>>>(hp, z, K2, 1);

  // ---- layer 3: conv ----
  zero_kernel<<<cdiv(N3 * HDIM, 256), 256, 0, stream>>>(agg, N3 * HDIM);
  aggH_kernel<<<cdiv(EDGES * 32, 256), 256, 0, stream>>>(hp, srcw, dstw, maskw, agg, EDGES);
  gemm_wmma_f32<true, true><<<gemm_grid(N3, HDIM), 256, 0, stream>>>(
      agg, hp, Wrel3, Wroot3, b3, h, N3, HDIM, HDIM);

  // ---- pool 3 (ratio 0.8: 205 -> 164) + readout x3 (edges no longer needed) ----
  score_kernel<<<cdiv(N3, 256), 256, 0, stream>>>(h, pw3, invn, 2, score, N3);
  topk_kernel<<<BGRAPH, 256, 0, stream>>>(score, perm, newid, K2, K3);
  pool_gather_kernel<<<cdiv(N4 * HDIM, 256), 256, 0, stream>>>(h, score, perm, hp, N4);
  readout_kernel<<<BGRAPH, 128, 0, stream>>>(hp, z, K3, 1);

  // ---- MLP head: 256 -> 512 relu -> 64 relu -> 1 ----
  gemm_wmma_f32<false, true><<<gemm_grid(BGRAPH, 512), 256, 0, stream>>>(
      z, nullptr, Wl1, nullptr, bl1, z1v, BGRAPH, 512, 256);
  gemm_wmma_f32<false, true><<<gemm_grid(BGRAPH, 64), 256, 0, stream>>>(
      z1v, nullptr, Wl2, nullptr, bl2, z2v, BGRAPH, 64, 512);
  mlp_final_kernel<<<cdiv(BGRAPH, 128), 128, 0, stream>>>(z2v, Wl3, bl3, out);
}